// DecoupledModel_2662879724147
// MI455X (gfx1250) — compile-verified
//
#include <hip/hip_runtime.h>
#include <hip/hip_bf16.h>

typedef float v2f __attribute__((ext_vector_type(2)));
typedef float v8f __attribute__((ext_vector_type(8)));

#define N_ENT   50000
#define N_EDGE  600000
#define FEAT    128
#define HID     256
#define OUT_DIM 128
#define BN_EPS  1e-5f

// Workspace layout (float offsets). Zero region is contiguous: [0, H_OFF).
#define AGG_OFF 0u
#define DEG_OFF 6400000u            // N_ENT*FEAT
#define SUM_OFF 6450000u            // + N_ENT
#define SSQ_OFF 6450256u            // + HID
#define H_OFF   6450512u            // + HID ; h/y buffer: N_ENT*HID floats

// f32 WMMA: D(16x16 f32) = A(16x4 f32) * B(4x16 f32) + C
// A layout: lanes 0-15 hold row M=lane, K={0,1}; lanes 16-31 hold K={2,3}.
// B layout (mirror): lanes 0-15 hold col N=lane, K={0,1}; lanes 16-31 K={2,3}.
// C/D layout: VGPR r -> M = r + (lane>=16 ? 8 : 0), N = lane & 15.
__device__ __forceinline__ v8f wmma4(v2f a, v2f b, v8f c) {
    return __builtin_amdgcn_wmma_f32_16x16x4_f32(
        /*neg_a=*/false, a, /*neg_b=*/false, b,
        /*c_mod=*/(short)0, c, /*reuse_a=*/false, /*reuse_b=*/false);
}

// ---------------------------------------------------------------- zero
__global__ void zero_kernel(float* __restrict__ p, unsigned long long n) {
    unsigned long long i = (unsigned long long)blockIdx.x * blockDim.x + threadIdx.x;
    unsigned long long stride = (unsigned long long)gridDim.x * blockDim.x;
    for (; i < n; i += stride) p[i] = 0.0f;
}

// ------------------------------------------------------------- scatter
// One wave per edge; lane handles 4 contiguous floats (128 = 32*4).
__global__ void __launch_bounds__(256) scatter_kernel(
    const int* __restrict__ edge_index, const int* __restrict__ edge_type,
    const float* __restrict__ x, const float* __restrict__ rel,
    float* __restrict__ agg, float* __restrict__ deg) {
    const int e    = (blockIdx.x * blockDim.x + threadIdx.x) >> 5;
    const int lane = threadIdx.x & 31;
    if (e >= N_EDGE) return;
    const int src = edge_index[e];
    const int dst = edge_index[N_EDGE + e];
    const int r   = edge_type[e];
    const float4 xv = *(const float4*)(x   + (size_t)src * FEAT + lane * 4);
    const float4 rv = *(const float4*)(rel + (size_t)r   * FEAT + lane * 4);
    float* a = agg + (size_t)dst * FEAT + lane * 4;
    atomicAdd(a + 0, xv.x + rv.x);
    atomicAdd(a + 1, xv.y + rv.y);
    atomicAdd(a + 2, xv.z + rv.z);
    atomicAdd(a + 3, xv.w + rv.w);
    if (lane == 0) atomicAdd(deg + dst, 1.0f);
}

// ------------------------------------------------------------- gemm1
// h[m][n] = sum_k Aeff[m][k] * Beff[k][n] + b_msg[n]
// Aeff = [agg/deg | x] (K = 256), Beff = [W_msg ; W_self].
// K-loop split at the 128 boundary -> two branch-free inner loops.
// Grid: 3125 blocks (16-row strip each); 8 waves x 32 cols = 256 cols.
__global__ void __launch_bounds__(256) gemm1_kernel(
    const float* __restrict__ x, const float* __restrict__ agg,
    const float* __restrict__ deg, const float* __restrict__ Wmsg,
    const float* __restrict__ Wself, const float* __restrict__ bmsg,
    float* __restrict__ h) {
    const int lane = threadIdx.x & 31;
    const int wave = threadIdx.x >> 5;
    const int half = lane >> 4;
    const int lrow = lane & 15;
    const int m    = blockIdx.x * 16 + lrow;
    const int n0   = wave * 32 + lrow;
    const int n1   = n0 + 16;
    const float invdeg = 1.0f / fmaxf(deg[m], 1.0f);

    const float* Arow0 = agg + (size_t)m * FEAT + half * 2;   // agg half
    const float* Arow1 = x   + (size_t)m * FEAT + half * 2;   // self half
    const float* B0    = Wmsg  + (size_t)(half * 2) * HID;
    const float* B1    = Wself + (size_t)(half * 2) * HID;

    v8f c0 = {}, c1 = {};
#pragma unroll 4
    for (int k = 0; k < FEAT; k += 4) {
        const float2 t = *(const float2*)(Arow0 + k);
        v2f a; a[0] = t.x * invdeg; a[1] = t.y * invdeg;
        const float* Brow = B0 + (size_t)k * HID;
        v2f b0, b1;
        b0[0] = Brow[n0]; b0[1] = Brow[HID + n0];
        b1[0] = Brow[n1]; b1[1] = Brow[HID + n1];
        c0 = wmma4(a, b0, c0);
        c1 = wmma4(a, b1, c1);
    }
#pragma unroll 4
    for (int k = 0; k < FEAT; k += 4) {
        const float2 t = *(const float2*)(Arow1 + k);
        v2f a; a[0] = t.x; a[1] = t.y;
        const float* Brow = B1 + (size_t)k * HID;
        v2f b0, b1;
        b0[0] = Brow[n0]; b0[1] = Brow[HID + n0];
        b1[0] = Brow[n1]; b1[1] = Brow[HID + n1];
        c0 = wmma4(a, b0, c0);
        c1 = wmma4(a, b1, c1);
    }
    const int mBase = blockIdx.x * 16 + half * 8;
    const float bb0 = bmsg[n0], bb1 = bmsg[n1];
#pragma unroll
    for (int r = 0; r < 8; ++r) {
        h[(size_t)(mBase + r) * HID + n0] = c0[r] + bb0;
        h[(size_t)(mBase + r) * HID + n1] = c1[r] + bb1;
    }
}

// ------------------------------------------------------------- gemm2
// y = h @ W1 + b1, written IN PLACE over h (block owns its 16-row strip;
// __syncthreads separates last K-read from first store). Also accumulates
// per-column sum / sum-of-squares for the batch-norm.
__global__ void __launch_bounds__(256) gemm2_kernel(
    float* __restrict__ hy, const float* __restrict__ W1,
    const float* __restrict__ b1, float* __restrict__ colsum,
    float* __restrict__ colssq) {
    const int lane = threadIdx.x & 31;
    const int wave = threadIdx.x >> 5;
    const int half = lane >> 4;
    const int lrow = lane & 15;
    const int m    = blockIdx.x * 16 + lrow;
    const int n0   = wave * 32 + lrow;
    const int n1   = n0 + 16;

    const float* Arow = hy + (size_t)m * HID + half * 2;
    const float* Bbase = W1 + (size_t)(half * 2) * HID;

    v8f c0 = {}, c1 = {};
#pragma unroll 4
    for (int k = 0; k < HID; k += 4) {
        const v2f a = *(const v2f*)(Arow + k);
        const float* Brow = Bbase + (size_t)k * HID;
        v2f b0, b1;
        b0[0] = Brow[n0]; b0[1] = Brow[HID + n0];
        b1[0] = Brow[n1]; b1[1] = Brow[HID + n1];
        c0 = wmma4(a, b0, c0);
        c1 = wmma4(a, b1, c1);
    }
    __syncthreads();   // all waves done reading the h strip before overwrite
    const int mBase = blockIdx.x * 16 + half * 8;
    const float bb0 = b1[n0], bb1 = b1[n1];
    float s0 = 0.f, q0 = 0.f, s1 = 0.f, q1 = 0.f;
#pragma unroll
    for (int r = 0; r < 8; ++r) {
        const float y0 = c0[r] + bb0;
        const float y1 = c1[r] + bb1;
        hy[(size_t)(mBase + r) * HID + n0] = y0;
        hy[(size_t)(mBase + r) * HID + n1] = y1;
        s0 += y0; q0 += y0 * y0;
        s1 += y1; q1 += y1 * y1;
    }
    atomicAdd(colsum + n0, s0); atomicAdd(colssq + n0, q0);
    atomicAdd(colsum + n1, s1); atomicAdd(colssq + n1, q1);
}

// ------------------------------------------------------- bn finalize
// Turn (sum, sumsq) into affine (scale, shift) in place: 256 threads.
__global__ void bn_finalize_kernel(float* __restrict__ colsum,
                                   float* __restrict__ colssq,
                                   const float* __restrict__ gamma,
                                   const float* __restrict__ beta) {
    const int k = threadIdx.x;
    const float mean = colsum[k] * (1.0f / N_ENT);
    const float var  = colssq[k] * (1.0f / N_ENT) - mean * mean;
    const float s    = gamma[k] * rsqrtf(var + BN_EPS);
    colsum[k] = s;                   // scale
    colssq[k] = beta[k] - mean * s;  // shift
}

// ------------------------------------------------------------- gemm3
// out = relu(y*scale + shift) @ W2 + b2 ; BN+ReLU folded into A-load.
// scale/shift staged in LDS (2 KB), read as float2 ds loads per K-step.
// 8 waves x 16 cols = 128 output cols.
__global__ void __launch_bounds__(256) gemm3_kernel(
    const float* __restrict__ y, const float* __restrict__ scale,
    const float* __restrict__ shift, const float* __restrict__ W2,
    const float* __restrict__ b2, float* __restrict__ out) {
    __shared__ float s_scale[HID];
    __shared__ float s_shift[HID];
    s_scale[threadIdx.x] = scale[threadIdx.x];
    s_shift[threadIdx.x] = shift[threadIdx.x];
    __syncthreads();

    const int lane = threadIdx.x & 31;
    const int wave = threadIdx.x >> 5;
    const int half = lane >> 4;
    const int lrow = lane & 15;
    const int m    = blockIdx.x * 16 + lrow;
    const int n    = wave * 16 + lrow;

    const float* Arow  = y  + (size_t)m * HID + half * 2;
    const float* Bbase = W2 + (size_t)(half * 2) * OUT_DIM;
    const float2* sc2  = (const float2*)(s_scale + half * 2);
    const float2* sh2  = (const float2*)(s_shift + half * 2);

    v8f c = {};
#pragma unroll 4
    for (int k = 0; k < HID; k += 4) {
        const float2 t  = *(const float2*)(Arow + k);
        const float2 sc = sc2[k >> 1];   // (s_scale+half*2)[k], [k+1]
        const float2 sh = sh2[k >> 1];
        v2f a;
        a[0] = fmaxf(0.0f, t.x * sc.x + sh.x);
        a[1] = fmaxf(0.0f, t.y * sc.y + sh.y);
        const float* Brow = Bbase + (size_t)k * OUT_DIM;
        v2f b;
        b[0] = Brow[n];
        b[1] = Brow[OUT_DIM + n];
        c = wmma4(a, b, c);
    }
    const int mBase = blockIdx.x * 16 + half * 8;
    const float bb = b2[n];
#pragma unroll
    for (int r = 0; r < 8; ++r)
        out[(size_t)(mBase + r) * OUT_DIM + n] = c[r] + bb;
}

// -------------------------------------------------------------- launch
extern "C" void kernel_launch(void* const* d_in, const int* in_sizes, int n_in,
                              void* d_out, int out_size, void* d_ws, size_t ws_size,
                              hipStream_t stream) {
    const int*   edge_index = (const int*)d_in[0];
    const int*   edge_type  = (const int*)d_in[1];
    const float* x          = (const float*)d_in[2];
    const float* rel        = (const float*)d_in[3];
    const float* Wmsg       = (const float*)d_in[4];
    const float* bmsg       = (const float*)d_in[5];
    const float* Wself      = (const float*)d_in[6];
    const float* W1         = (const float*)d_in[7];
    const float* b1v        = (const float*)d_in[8];
    const float* gamma      = (const float*)d_in[9];
    const float* beta       = (const float*)d_in[10];
    const float* W2         = (const float*)d_in[11];
    const float* b2v        = (const float*)d_in[12];

    float* ws     = (float*)d_ws;
    float* agg    = ws + AGG_OFF;
    float* deg    = ws + DEG_OFF;
    float* colsum = ws + SUM_OFF;
    float* colssq = ws + SSQ_OFF;
    float* h      = ws + H_OFF;
    float* out    = (float*)d_out;

    // 1) zero accumulators (agg, deg, bn sums) — required every call.
    zero_kernel<<<4096, 256, 0, stream>>>(ws, (unsigned long long)H_OFF);
    // 2) edge scatter: one wave per edge.
    scatter_kernel<<<(N_EDGE * 32 + 255) / 256, 256, 0, stream>>>(
        edge_index, edge_type, x, rel, agg, deg);
    // 3) h = [agg/deg | x] @ [W_msg ; W_self] + b_msg
    gemm1_kernel<<<N_ENT / 16, 256, 0, stream>>>(x, agg, deg, Wmsg, Wself, bmsg, h);
    // 4) y = h @ W1 + b1 (in place) + column stats
    gemm2_kernel<<<N_ENT / 16, 256, 0, stream>>>(h, W1, b1v, colsum, colssq);
    // 5) batch-norm scale/shift
    bn_finalize_kernel<<<1, HID, 0, stream>>>(colsum, colssq, gamma, beta);
    // 6) out = relu(bn(y)) @ W2 + b2
    gemm3_kernel<<<N_ENT / 16, 256, 0, stream>>>(h, colsum, colssq, W2, b2v, out);
}